// MemoryBank_60352880443930
// MI455X (gfx1250) — compile-verified
//
#include <hip/hip_runtime.h>
#include <cstdint>
#include <cstddef>

typedef __attribute__((ext_vector_type(2))) float v2f;
typedef __attribute__((ext_vector_type(8))) float v8f;

#define T_TOK   32768
#define K_IDX   8
#define N_SLOT  128
#define D_DIM   4096
#define B_BATCH 4
#define TSPLIT  8
#define TCHUNK  (T_TOK / TSPLIT)   // 4096 tokens per grid.y slice

#define EMA_ALPHA 0.1f

// workspace layout (bytes)
#define WS_SUMS_BYTES ((size_t)N_SLOT * D_DIM * sizeof(float))   // 2 MiB
#define WS_CNT_OFF    WS_SUMS_BYTES
#define WS_MASK_OFF   (WS_CNT_OFF + 1024)                        // keep 16B align
// masks: T_TOK * 16B = 512 KiB  -> total ~2.6 MiB

// ---------------------------------------------------------------------------
// Kernel 1: build 128-bit touch mask per token (dedups K duplicates via OR)
//           and accumulate per-slot token counts.
// ---------------------------------------------------------------------------
__global__ void __launch_bounds__(256)
mask_build_kernel(const long long* __restrict__ slots,
                  uint4* __restrict__ maskbits,
                  unsigned* __restrict__ counts) {
    __shared__ unsigned lcnt[N_SLOT];
    const int tid = threadIdx.x;
    if (tid < N_SLOT) lcnt[tid] = 0u;
    __syncthreads();

    const int t = blockIdx.x * blockDim.x + tid;
    if (t < T_TOK) {
        unsigned w[4] = {0u, 0u, 0u, 0u};
#pragma unroll
        for (int j = 0; j < K_IDX; ++j) {
            int n = (int)slots[(size_t)t * K_IDX + j] & (N_SLOT - 1);
            w[n >> 5] |= 1u << (n & 31);
        }
        uint4 v; v.x = w[0]; v.y = w[1]; v.z = w[2]; v.w = w[3];
        maskbits[t] = v;
#pragma unroll
        for (int wi = 0; wi < 4; ++wi) {
            unsigned b = w[wi];
            while (b) {
                int k = __ffs(b) - 1;
                atomicAdd(&lcnt[wi * 32 + k], 1u);
                b &= b - 1u;
            }
        }
    }
    __syncthreads();
    if (tid < N_SLOT && lcnt[tid] != 0u) atomicAdd(&counts[tid], lcnt[tid]);
}

// ---------------------------------------------------------------------------
// Kernel 2: sums[N=128, D=4096] += m^T (128 x Tchunk) @ H (Tchunk x 4096)
// One wave per 16-wide D tile; holds all 8 N-tiles (16x16 f32 accumulators)
// so H is streamed from HBM exactly once. Full-f32 WMMA (16x16x4).
//
// f32 A (16x4) layout: lanes 0-15 -> M=0..15, lanes 16-31 -> M=0..15
//   VGPR0 = K0 (low half) / K2 (high half), VGPR1 = K1 / K3
// f32 B (4x16): mirrored -> VGPR0 = row K0 / K2, VGPR1 = row K1 / K3, N=lane&15
// f32 C/D (16x16): VGPR v -> M = v + 8*(lane>=16), N = lane&15
// ---------------------------------------------------------------------------
__global__ void __launch_bounds__(256)
slot_gemm_kernel(const float* __restrict__ H,
                 const uint4* __restrict__ maskbits,
                 float* __restrict__ sums) {
    const int lane = threadIdx.x & 31;
    const int wave = threadIdx.x >> 5;
    const int m    = lane & 15;          // column within tile / M row for A
    const int h    = lane >> 4;          // half-wave: 0 or 1
    const int k0   = h * 2;              // K offset carried by this half

    const int d0   = (blockIdx.x * 8 + wave) * 16;   // D tile base
    const int tbeg = blockIdx.y * TCHUNK;

    v8f acc[8];
#pragma unroll
    for (int i = 0; i < 8; ++i) acc[i] = (v8f){0.f,0.f,0.f,0.f,0.f,0.f,0.f,0.f};

    const float* Hb = H + (size_t)d0 + m;

    for (int t = tbeg; t < tbeg + TCHUNK; t += 4) {
        // token bitmasks for the two K slots this half-wave supplies
        uint4 mk0 = maskbits[t + k0];
        uint4 mk1 = maskbits[t + k0 + 1];
        unsigned mw0[4] = {mk0.x, mk0.y, mk0.z, mk0.w};
        unsigned mw1[4] = {mk1.x, mk1.y, mk1.z, mk1.w};

        // B tile: H rows t+k0 and t+k0+1, columns d0..d0+15 (coalesced b32)
        v2f b;
        b.x = Hb[(size_t)(t + k0)     * D_DIM];
        b.y = Hb[(size_t)(t + k0 + 1) * D_DIM];

#pragma unroll
        for (int i = 0; i < 8; ++i) {                 // 8 N-tiles (slots 16i..16i+15)
            const unsigned sh = (unsigned)(((i & 1) << 4) + m);
            v2f a;
            a.x = (float)((mw0[i >> 1] >> sh) & 1u);
            a.y = (float)((mw1[i >> 1] >> sh) & 1u);
            acc[i] = __builtin_amdgcn_wmma_f32_16x16x4_f32(
                /*neg_a=*/false, a, /*neg_b=*/false, b,
                /*c_mod=*/(short)0, acc[i],
                /*reuse_a=*/false, /*reuse_b=*/false);
        }
    }

    // merge partial tiles across T slices with HW fp32 atomics
#pragma unroll
    for (int i = 0; i < 8; ++i) {
#pragma unroll
        for (int v = 0; v < 8; ++v) {
            const int row = 16 * i + v + 8 * h;       // slot index
            unsafeAtomicAdd(&sums[(size_t)row * D_DIM + d0 + m], acc[i][v]);
        }
    }
}

// ---------------------------------------------------------------------------
// Kernel 3: out = memory, with row batch_idx EMA-updated where counts > 0
// ---------------------------------------------------------------------------
__global__ void __launch_bounds__(256)
finalize_kernel(const float* __restrict__ mem,
                const float* __restrict__ sums,
                const unsigned* __restrict__ counts,
                const int* __restrict__ batch_ptr,
                float* __restrict__ out) {
    const size_t idx   = (size_t)blockIdx.x * blockDim.x + threadIdx.x;
    const size_t total = (size_t)B_BATCH * N_SLOT * D_DIM;
    if (idx >= total) return;

    const float cur = mem[idx];
    float r = cur;

    const int    batch = batch_ptr[0];
    const size_t base  = (size_t)batch * N_SLOT * D_DIM;
    if (idx >= base && idx < base + (size_t)N_SLOT * D_DIM) {
        const size_t rel = idx - base;
        const int    n   = (int)(rel >> 12);          // rel / D_DIM
        const unsigned c = counts[n];
        if (c > 0u) {
            const float agg = sums[rel] / fmaxf((float)c, 1.0f);
            r = EMA_ALPHA * agg + (1.0f - EMA_ALPHA) * cur;
        }
    }
    out[idx] = r;
}

// ---------------------------------------------------------------------------
extern "C" void kernel_launch(void* const* d_in, const int* in_sizes, int n_in,
                              void* d_out, int out_size, void* d_ws, size_t ws_size,
                              hipStream_t stream) {
    const float*     memory = (const float*)d_in[0];       // [4,128,4096] f32
    const float*     hidden = (const float*)d_in[1];       // [32768,4096] f32
    const long long* slots  = (const long long*)d_in[2];   // [32768,8] int64
    const int*       bptr   = (const int*)d_in[3];         // scalar batch_idx

    float*    sums   = (float*)d_ws;
    unsigned* counts = (unsigned*)((char*)d_ws + WS_CNT_OFF);
    uint4*    maskb  = (uint4*)((char*)d_ws + WS_MASK_OFF);
    float*    out    = (float*)d_out;

    // zero sums + counts (replay-deterministic; capturable memset node)
    hipMemsetAsync(d_ws, 0, WS_MASK_OFF, stream);

    mask_build_kernel<<<T_TOK / 256, 256, 0, stream>>>(slots, maskb, counts);

    slot_gemm_kernel<<<dim3(D_DIM / (16 * 8), TSPLIT), 256, 0, stream>>>(
        hidden, maskb, sums);

    const size_t total = (size_t)B_BATCH * N_SLOT * D_DIM;
    finalize_kernel<<<(unsigned)((total + 255) / 256), 256, 0, stream>>>(
        memory, sums, counts, bptr, out);
}